// LocalGlobalTokenPartialMemoryLM_34368328303152
// MI455X (gfx1250) — compile-verified
//
#include <hip/hip_runtime.h>
#include <math.h>

// ---------------------------------------------------------------------------
// Model dims (fixed by the reference)
// ---------------------------------------------------------------------------
#define DIM_V  32000
#define DIM_E  256
#define DIM_H  512
#define DIM_M  128
#define DIM_U  4096
#define DIM_B  2
#define DIM_S  512
#define DIM_LW 64
#define DIM_CS 64
#define DIM_C  8          // ceil(S/CS)
#define NEGF   (-3.4028235e38f)

typedef __attribute__((ext_vector_type(2))) float v2f;
typedef __attribute__((ext_vector_type(8))) float v8f;

// V_WMMA_F32_16X16X4_F32 : D(16x16 f32) = A(16x4 f32) * B(4x16 f32) + C
__device__ __forceinline__ v8f wmma4(v2f a, v2f b, v8f c) {
    return __builtin_amdgcn_wmma_f32_16x16x4_f32(
        /*neg_a=*/false, a, /*neg_b=*/false, b,
        /*c_mod=*/(short)0, c, /*reuse_a=*/false, /*reuse_b=*/false);
}

// ---------------------------------------------------------------------------
// 1) embedding gather: emb[bs,e] = embedding[input_ids[bs], e]
// ---------------------------------------------------------------------------
__global__ void k_gather_emb(const int* __restrict__ ids,
                             const float* __restrict__ embedding,
                             float* __restrict__ emb, int total) {
    int idx = blockIdx.x * blockDim.x + threadIdx.x;
    if (idx >= total) return;
    int bs = idx / DIM_E, e = idx % DIM_E;
    emb[idx] = embedding[(size_t)ids[bs] * DIM_E + e];
}

// ---------------------------------------------------------------------------
// Generic wave-per-tile-group WMMA GEMM (NB N-tiles per wave for A reuse):
//   Y[Mrows,N] = act( X[Mrows,K] @ W[N,K]^T + bias )
// A lanes: 0-15 -> M=lane, K-pair (k0,k0+1); 16-31 -> M=lane-16, K-pair (k0+2,k0+3)
// B lanes: 0-15 -> N=lane, same K pairing.  D: vgpr r -> M = r + 8*(lane>=16), N = lane&15.
// Requires N % (16*NB) == 0 (all model Ns are multiples of 64).
// ---------------------------------------------------------------------------
template <int ACT, int NB>   // ACT: 0 = identity, 1 = square(relu(x))
__global__ void k_gemm_xwT(const float* __restrict__ X, const float* __restrict__ W,
                           const float* __restrict__ bias, float* __restrict__ Y,
                           int Mrows, int N, int K) {
    int gtid = blockIdx.x * blockDim.x + threadIdx.x;
    int wave = gtid >> 5;
    int lane = gtid & 31;
    int ngroups = (N / 16) / NB;
    int mtiles  = (Mrows + 15) >> 4;
    if (wave >= ngroups * mtiles) return;
    int mt = wave / ngroups, ng = wave % ngroups;
    int half = lane >> 4;            // 0: lanes 0-15, 1: lanes 16-31
    int l    = lane & 15;
    int ra = mt * 16 + l; if (ra >= Mrows) ra = Mrows - 1;   // clamped A row
    const float* arow = X + (size_t)ra * K;
    const float* brow[NB];
#pragma unroll
    for (int j = 0; j < NB; ++j)
        brow[j] = W + (size_t)((ng * NB + j) * 16 + l) * K;
    v8f acc[NB];
    v8f zero = {};
#pragma unroll
    for (int j = 0; j < NB; ++j) acc[j] = zero;
    for (int k0 = 0; k0 < K; k0 += 4) {
        int ka = k0 + 2 * half;
        v2f a; a.x = arow[ka]; a.y = arow[ka + 1];
#pragma unroll
        for (int j = 0; j < NB; ++j) {
            v2f b; b.x = brow[j][ka]; b.y = brow[j][ka + 1];
            acc[j] = wmma4(a, b, acc[j]);      // NB independent chains, shared A
        }
    }
#pragma unroll
    for (int j = 0; j < NB; ++j) {
        union { v8f v; float f[8]; } u; u.v = acc[j];
        int col = (ng * NB + j) * 16 + l;
        float bv = bias ? bias[col] : 0.f;
#pragma unroll
        for (int r = 0; r < 8; ++r) {
            int row = mt * 16 + r + 8 * half;
            if (row >= Mrows) continue;
            float val = u.f[r] + bv;
            if (ACT == 1) { val = fmaxf(val, 0.f); val = val * val; }
            Y[(size_t)row * N + col] = val;
        }
    }
}

// ---------------------------------------------------------------------------
// 3) GRU scan. One block per batch, 1024 threads: each hidden unit's K=512
//    dot is split across two threads (halves of h), combined through LDS.
// ---------------------------------------------------------------------------
__global__ void k_gru_scan(const float* __restrict__ xg,
                           const float* __restrict__ w_hh,
                           const float* __restrict__ b_hh,
                           float* __restrict__ states) {
    __shared__ float h[DIM_H];
    __shared__ float pr[DIM_H], pz[DIM_H], pn[DIM_H];
    int b   = blockIdx.x;
    int t0  = threadIdx.x;               // 0..1023
    int j   = t0 & (DIM_H - 1);          // hidden unit
    int seg = t0 >> 9;                   // 0: e in [0,256), 1: e in [256,512)
    int e0  = seg * (DIM_H / 2);
    if (t0 < DIM_H) h[t0] = 0.f;
    __syncthreads();
    const float* wr = w_hh + (size_t)j * DIM_H + e0;
    const float* wz = w_hh + (size_t)(DIM_H + j) * DIM_H + e0;
    const float* wn = w_hh + (size_t)(2 * DIM_H + j) * DIM_H + e0;
    for (int t = 0; t < DIM_S; ++t) {
        float hr = 0.f, hz = 0.f, hn = 0.f;
        for (int e = 0; e < DIM_H / 2; ++e) {
            float hv = h[e0 + e];
            hr = fmaf(wr[e], hv, hr);
            hz = fmaf(wz[e], hv, hz);
            hn = fmaf(wn[e], hv, hn);
        }
        if (seg) { pr[j] = hr; pz[j] = hz; pn[j] = hn; }
        __syncthreads();
        if (!seg) {
            hr += pr[j] + b_hh[j];
            hz += pz[j] + b_hh[DIM_H + j];
            hn += pn[j] + b_hh[2 * DIM_H + j];
            const float* xt = xg + ((size_t)b * DIM_S + t) * (3 * DIM_H);
            float r = 1.f / (1.f + expf(-(xt[j] + hr)));
            float z = 1.f / (1.f + expf(-(xt[DIM_H + j] + hz)));
            float c = tanhf(xt[2 * DIM_H + j] + r * hn);
            float hnew = (1.f - z) * c + z * h[j];
            h[j] = hnew;
            states[((size_t)b * DIM_S + t) * DIM_H + j] = hnew;
        }
        __syncthreads();
    }
}

// ---------------------------------------------------------------------------
// 10) chunk summary: masked mean over each CS chunk of states
// ---------------------------------------------------------------------------
__global__ void k_summary(const float* __restrict__ states, float* __restrict__ summary) {
    int bc = blockIdx.x;            // b*C + c
    int b = bc / DIM_C, c = bc % DIM_C;
    int j = threadIdx.x;            // 0..H-1
    float s = 0.f; int cnt = 0;
    for (int i = 0; i < DIM_CS; ++i) {
        int t = c * DIM_CS + i;
        if (t < DIM_S) { s += states[((size_t)b * DIM_S + t) * DIM_H + j]; ++cnt; }
    }
    summary[(size_t)bc * DIM_H + j] = s / (float)max(cnt, 1);
}

// ---------------------------------------------------------------------------
// 13) mixture softmax -> per-row scales
// ---------------------------------------------------------------------------
__global__ void k_mix(const float* __restrict__ states,
                      const float* __restrict__ mix_w, const float* __restrict__ mix_b,
                      const float* __restrict__ lsc, const float* __restrict__ gsc,
                      float* __restrict__ lscale, float* __restrict__ gscale, int BS) {
    int bs = blockIdx.x * blockDim.x + threadIdx.x;
    if (bs >= BS) return;
    const float* st = states + (size_t)bs * DIM_H;
    float m0 = mix_b[0], m1 = mix_b[1];
    for (int e = 0; e < DIM_H; ++e) {
        m0 = fmaf(st[e], mix_w[e], m0);
        m1 = fmaf(st[e], mix_w[DIM_H + e], m1);
    }
    float mx = fmaxf(m0, m1);
    float e0 = expf(m0 - mx), e1 = expf(m1 - mx);
    float inv = 1.f / (e0 + e1);
    lscale[bs] = e0 * inv * lsc[0];
    gscale[bs] = e1 * inv * gsc[0];
}

// ---------------------------------------------------------------------------
// 14) global chunk attention -> ctx[b,s,E]
// ---------------------------------------------------------------------------
__global__ void k_global_ctx(const float* __restrict__ gq, const float* __restrict__ gk,
                             const float* __restrict__ gv, float* __restrict__ ctx) {
    __shared__ float a[DIM_C];
    int bq = blockIdx.x;
    int b = bq / DIM_S, s = bq % DIM_S;
    int tid = threadIdx.x;          // 256 threads
    if (tid < DIM_C) {
        int chunk_end = min((tid + 1) * DIM_CS - 1, DIM_S - 1);
        float d = NEGF;
        if (chunk_end < s - DIM_LW) {
            const float* qp = gq + ((size_t)b * DIM_S + s) * DIM_M;
            const float* kp = gk + ((size_t)b * DIM_C + tid) * DIM_M;
            float acc = 0.f;
            for (int m = 0; m < DIM_M; ++m) acc = fmaf(qp[m], kp[m], acc);
            d = acc * (1.f / sqrtf((float)DIM_M));
        }
        a[tid] = d;
    }
    __syncthreads();
    if (tid == 0) {
        float mx = NEGF;
        for (int c = 0; c < DIM_C; ++c) mx = fmaxf(mx, a[c]);
        float tmp[DIM_C]; float ssum = 0.f;
        for (int c = 0; c < DIM_C; ++c) {
            float e = (a[c] > -1e38f) ? expf(a[c] - mx) : 0.f;
            tmp[c] = e; ssum += e;
        }
        float inv = 1.f / fmaxf(ssum, 1e-6f);
        for (int c = 0; c < DIM_C; ++c) a[c] = tmp[c] * inv;
    }
    __syncthreads();
    if (tid < DIM_E) {
        float acc = 0.f;
        for (int c = 0; c < DIM_C; ++c)
            acc = fmaf(a[c], gv[((size_t)b * DIM_C + c) * DIM_E + tid], acc);
        ctx[((size_t)b * DIM_S + s) * DIM_E + tid] = acc;
    }
}

// ---------------------------------------------------------------------------
// 15) local windowed attention + token scatter (atomic add into d_out)
// ---------------------------------------------------------------------------
__global__ void k_local_attn(const float* __restrict__ q, const float* __restrict__ k,
                             const int* __restrict__ ids, const float* __restrict__ lscale,
                             float* __restrict__ out) {
    __shared__ float red[DIM_LW];
    int bq = blockIdx.x;
    int b = bq / DIM_S, qpos = bq % DIM_S;
    int tid = threadIdx.x;                 // 0..LW-1
    int kpos = qpos - DIM_LW + tid;        // keys in [qpos-LW, qpos-1]
    bool valid = (kpos >= 0) && (kpos < qpos);
    float sc = NEGF;
    if (valid) {
        const float* qp = q + ((size_t)b * DIM_S + qpos) * DIM_M;
        const float* kp = k + ((size_t)b * DIM_S + kpos) * DIM_M;
        float d = 0.f;
        for (int m = 0; m < DIM_M; ++m) d = fmaf(qp[m], kp[m], d);
        sc = d * (1.f / sqrtf((float)DIM_M));
    }
    red[tid] = sc; __syncthreads();
    for (int o = DIM_LW / 2; o > 0; o >>= 1) {
        if (tid < o) red[tid] = fmaxf(red[tid], red[tid + o]);
        __syncthreads();
    }
    float mx = red[0]; __syncthreads();
    float ex = valid ? expf(sc - mx) : 0.f;
    red[tid] = ex; __syncthreads();
    for (int o = DIM_LW / 2; o > 0; o >>= 1) {
        if (tid < o) red[tid] += red[tid + o];
        __syncthreads();
    }
    float ssum = red[0];
    float w = ex / fmaxf(ssum, 1e-6f);
    if (valid && w != 0.f) {
        float scale = lscale[(size_t)b * DIM_S + qpos];
        atomicAdd(&out[((size_t)b * DIM_S + qpos) * DIM_V + ids[(size_t)b * DIM_S + kpos]],
                  scale * w);
    }
}

// ---------------------------------------------------------------------------
// 16) dual WMMA tile group (4 N-tiles, A reused): partial = feat@pw^T,
//     gpart = ctx@gpw^T; scatter-add into out at untied columns.
// ---------------------------------------------------------------------------
__global__ void k_partial_scatter(const float* __restrict__ feat, const float* __restrict__ ctx,
                                  const float* __restrict__ pw, const float* __restrict__ pb,
                                  const float* __restrict__ gpw, const int* __restrict__ untied,
                                  const float* __restrict__ gscale, float* __restrict__ out,
                                  int Mrows) {
    constexpr int NB = 4;
    int gtid = blockIdx.x * blockDim.x + threadIdx.x;
    int wave = gtid >> 5;
    int lane = gtid & 31;
    int ngroups = (DIM_U / 16) / NB;
    if (wave >= (Mrows >> 4) * ngroups) return;
    int mt = wave / ngroups, ng = wave % ngroups;
    int half = lane >> 4, l = lane & 15;
    const float* arow  = feat + (size_t)(mt * 16 + l) * DIM_E;
    const float* a2row = ctx  + (size_t)(mt * 16 + l) * DIM_E;
    const float* brow[NB];
    const float* b2row[NB];
#pragma unroll
    for (int j = 0; j < NB; ++j) {
        size_t wroff = (size_t)((ng * NB + j) * 16 + l) * DIM_E;
        brow[j]  = pw  + wroff;
        b2row[j] = gpw + wroff;
    }
    v8f accP[NB], accG[NB];
    v8f zero = {};
#pragma unroll
    for (int j = 0; j < NB; ++j) { accP[j] = zero; accG[j] = zero; }
    for (int k0 = 0; k0 < DIM_E; k0 += 4) {
        int ka = k0 + 2 * half;
        v2f a, a2;
        a.x  = arow[ka];  a.y  = arow[ka + 1];
        a2.x = a2row[ka]; a2.y = a2row[ka + 1];
#pragma unroll
        for (int j = 0; j < NB; ++j) {
            v2f b, b2;
            b.x  = brow[j][ka];  b.y  = brow[j][ka + 1];
            b2.x = b2row[j][ka]; b2.y = b2row[j][ka + 1];
            accP[j] = wmma4(a,  b,  accP[j]);
            accG[j] = wmma4(a2, b2, accG[j]);
        }
    }
#pragma unroll
    for (int j = 0; j < NB; ++j) {
        union { v8f v; float f[8]; } uP, uG; uP.v = accP[j]; uG.v = accG[j];
        int col = (ng * NB + j) * 16 + l;
        int uid = untied[col];
        float pbias = pb[col];
#pragma unroll
        for (int r = 0; r < 8; ++r) {
            int row = mt * 16 + r + 8 * half;
            float val = uP.f[r] + pbias + gscale[row] * uG.f[r];
            atomicAdd(&out[(size_t)row * DIM_V + uid], val);
        }
    }
}

// ---------------------------------------------------------------------------
// host-side helpers / launcher
// ---------------------------------------------------------------------------
static void launch_gemm(int act, const float* X, const float* W, const float* bias,
                        float* Y, int Mrows, int N, int K, hipStream_t stream) {
    constexpr int NB = 4;                        // all model Ns are multiples of 64
    int waves = ((Mrows + 15) / 16) * ((N / 16) / NB);
    int threads = 128;                           // 4 waves / block
    int blocks = (waves * 32 + threads - 1) / threads;
    if (act)
        k_gemm_xwT<1, NB><<<blocks, threads, 0, stream>>>(X, W, bias, Y, Mrows, N, K);
    else
        k_gemm_xwT<0, NB><<<blocks, threads, 0, stream>>>(X, W, bias, Y, Mrows, N, K);
}

extern "C" void kernel_launch(void* const* d_in, const int* in_sizes, int n_in,
                              void* d_out, int out_size, void* d_ws, size_t ws_size,
                              hipStream_t stream) {
    const int BS = DIM_B * DIM_S;   // 1024

    const int*   input_ids = (const int*)d_in[0];
    const int*   untied    = (const int*)d_in[1];
    const float* embedding = (const float*)d_in[2];
    const float* w_ih      = (const float*)d_in[3];
    const float* w_hh      = (const float*)d_in[4];
    const float* b_ih      = (const float*)d_in[5];
    const float* b_hh      = (const float*)d_in[6];
    const float* fc_w      = (const float*)d_in[7];
    const float* fc_b      = (const float*)d_in[8];
    const float* proj_w    = (const float*)d_in[9];
    const float* proj_b    = (const float*)d_in[10];
    const float* out_bias  = (const float*)d_in[11];
    const float* partial_w = (const float*)d_in[12];
    const float* partial_b = (const float*)d_in[13];
    const float* lq_w = (const float*)d_in[14]; const float* lq_b = (const float*)d_in[15];
    const float* lk_w = (const float*)d_in[16]; const float* lk_b = (const float*)d_in[17];
    const float* gq_w = (const float*)d_in[18]; const float* gq_b = (const float*)d_in[19];
    const float* gk_w = (const float*)d_in[20]; const float* gk_b = (const float*)d_in[21];
    const float* gv_w = (const float*)d_in[22]; const float* gv_b = (const float*)d_in[23];
    const float* gpartial_w = (const float*)d_in[24];
    const float* mix_w = (const float*)d_in[25]; const float* mix_b = (const float*)d_in[26];
    const float* lsc = (const float*)d_in[27];   const float* gsc = (const float*)d_in[28];
    float* out = (float*)d_out;

    // workspace carve-up (floats)
    float* ws      = (float*)d_ws;
    float* emb     = ws; ws += (size_t)BS * DIM_E;
    float* xg      = ws; ws += (size_t)BS * 3 * DIM_H;
    float* states  = ws; ws += (size_t)BS * DIM_H;
    float* hf      = ws; ws += (size_t)BS * 4 * DIM_E;
    float* feat    = ws; ws += (size_t)BS * DIM_E;
    float* qb      = ws; ws += (size_t)BS * DIM_M;
    float* kb      = ws; ws += (size_t)BS * DIM_M;
    float* gqb     = ws; ws += (size_t)BS * DIM_M;
    float* summary = ws; ws += (size_t)DIM_B * DIM_C * DIM_H;
    float* gkb     = ws; ws += (size_t)DIM_B * DIM_C * DIM_M;
    float* gvb     = ws; ws += (size_t)DIM_B * DIM_C * DIM_E;
    float* ctx     = ws; ws += (size_t)BS * DIM_E;
    float* lscale  = ws; ws += BS;
    float* gscale  = ws; ws += BS;

    // 1) embedding gather
    {
        int total = BS * DIM_E;
        k_gather_emb<<<(total + 255) / 256, 256, 0, stream>>>(input_ids, embedding, emb, total);
    }
    // 2) xg = emb @ W_ih^T + b_ih                     [1024 x 1536], K=256
    launch_gemm(0, emb, w_ih, b_ih, xg, BS, 3 * DIM_H, DIM_E, stream);
    // 3) GRU scan -> states (1024 thr: split-K over two threads per unit)
    k_gru_scan<<<DIM_B, 2 * DIM_H, 0, stream>>>(xg, w_hh, b_hh, states);
    // 4) hf = square(relu(states @ fc_w^T + fc_b))    [1024 x 1024], K=512
    launch_gemm(1, states, fc_w, fc_b, hf, BS, 4 * DIM_E, DIM_H, stream);
    // 5) feat = hf @ proj_w^T + proj_b                [1024 x 256], K=1024
    launch_gemm(0, hf, proj_w, proj_b, feat, BS, DIM_E, 4 * DIM_E, stream);
    // 6) base logits -> out = feat @ embedding^T + output_bias   [1024 x 32000], K=256
    launch_gemm(0, feat, embedding, out_bias, out, BS, DIM_V, DIM_E, stream);
    // 7-9) q, k, gq projections                       [1024 x 128], K=512
    launch_gemm(0, states, lq_w, lq_b, qb,  BS, DIM_M, DIM_H, stream);
    launch_gemm(0, states, lk_w, lk_b, kb,  BS, DIM_M, DIM_H, stream);
    launch_gemm(0, states, gq_w, gq_b, gqb, BS, DIM_M, DIM_H, stream);
    // 10) chunk summaries
    k_summary<<<DIM_B * DIM_C, DIM_H, 0, stream>>>(states, summary);
    // 11-12) gk, gv                                   [16 x 128/256], K=512
    launch_gemm(0, summary, gk_w, gk_b, gkb, DIM_B * DIM_C, DIM_M, DIM_H, stream);
    launch_gemm(0, summary, gv_w, gv_b, gvb, DIM_B * DIM_C, DIM_E, DIM_H, stream);
    // 13) mixture scales
    k_mix<<<(BS + 255) / 256, 256, 0, stream>>>(states, mix_w, mix_b, lsc, gsc,
                                                lscale, gscale, BS);
    // 14) global context
    k_global_ctx<<<BS, 256, 0, stream>>>(gqb, gkb, gvb, ctx);
    // 15) local attention scatter into out
    k_local_attn<<<BS, DIM_LW, 0, stream>>>(qb, kb, input_ids, lscale, out);
    // 16) partial + global-partial dual-WMMA (4 N-tiles/wave) scatter into out
    {
        int waves = (BS / 16) * ((DIM_U / 16) / 4);
        int threads = 128;
        int blocks = (waves * 32 + threads - 1) / threads;
        k_partial_scatter<<<blocks, threads, 0, stream>>>(feat, ctx, partial_w, partial_b,
                                                          gpartial_w, untied, gscale, out, BS);
    }
}